// BCE_35270271435550
// MI455X (gfx1250) — compile-verified
//
#include <hip/hip_runtime.h>
#include <math.h>

typedef __attribute__((ext_vector_type(2))) float v2f;
typedef __attribute__((ext_vector_type(8))) float v8f;

#define DIM   64
#define HWSZ  4096       // 64*64
#define VOX   262144     // 64^3
#define NIMG  4
#define BVOX  1048576    // NIMG*VOX
#define CCH   32
#define PADF  4608       // > HWSZ + DIM + corner slack; guards unmasked loads

// LDS tile for conv2: 3 d-planes x 18 h-rows x 66 w-cols (halo zero-padded)
#define XS    66
#define PS    (18 * XS)      // 1188
#define TILEF (3 * PS)       // 3564 floats = 14256 B

// workspace layout (float offsets)
#define WS_FUZZ 0                        // BVOX + 2*PADF (fuzz at WS_FUZZ+PADF)
#define WS_P1   (BVOX + 2 * PADF)        // 4096 blocks * 2
#define WS_BN1  (WS_P1 + 8192)           // 2 floats (scale, shift)
#define WS_P2   (WS_BN1 + 4)             // 1024 blocks * 64
#define WS_BN2  (WS_P2 + 65536)          // 64 floats (scale[32], shift[32])

// ---------------------------------------------------------------------------
// K1: conv1 (32ch -> 1ch, 3x3x3, SAME) + bias + fuzzy Gaussian membership.
// Branchless: per-tap wrapped unsigned byte offsets + float validity masks
// precomputed once; channel loop is pure load/mask/fma, 27 loads in flight.
// ---------------------------------------------------------------------------
__global__ __launch_bounds__(256) void k1_conv1_fuzz(
    const float* __restrict__ x, const float* __restrict__ w1,
    const float* __restrict__ cb1, const float* __restrict__ mu,
    const float* __restrict__ sg, float* __restrict__ fuzz,
    float* __restrict__ part)
{
    const int t = blockIdx.x * 256 + threadIdx.x;        // global voxel id
    const int w = t & 63, h = (t >> 6) & 63, d = (t >> 12) & 63, img = t >> 18;
    const char* xb = (const char*)(x + (size_t)img * (CCH * VOX));

    unsigned off[27];   // wrapped byte offsets (always in-bounds, >= 0)
    float    msk[27];   // 1.0f if tap in-bounds else 0.0f
    #pragma unroll
    for (int tap = 0; tap < 27; ++tap) {
        const int dz = tap / 9 - 1;                      // compile-time
        const int dy = (tap / 3) % 3 - 1;
        const int dx = tap % 3 - 1;
        const int dd = d + dz, hh = h + dy, ww = w + dx;
        const bool v = ((unsigned)dd < 64u) & ((unsigned)hh < 64u) &
                       ((unsigned)ww < 64u);
        off[tap] = (unsigned)((((dd & 63) * HWSZ) + ((hh & 63) * DIM) +
                               (ww & 63)) * 4);
        msk[tap] = v ? 1.f : 0.f;
    }

    float acc = cb1[0];
    for (int c = 0; c < CCH; ++c) {
        const char* xc = xb + (size_t)c * (VOX * 4);
        const float* wc = w1 + c * 27;                   // uniform -> scalar
        #pragma unroll
        for (int tap = 0; tap < 27; ++tap) {
            const float v = *(const float*)(xc + off[tap]);
            acc = fmaf(v * msk[tap], wc[tap], acc);
        }
    }

    float ss = 0.f;
    #pragma unroll
    for (int n = 0; n < 4; ++n) {
        const float z = (acc - mu[n]) / sg[n];
        ss = fmaf(z, z, ss);
    }
    const float fz = __expf(-ss);
    fuzz[t] = fz;                                        // fuzz already +PADF

    __shared__ float sS[256], sQ[256];
    sS[threadIdx.x] = fz;
    sQ[threadIdx.x] = fz * fz;
    __syncthreads();
    for (int st = 128; st > 0; st >>= 1) {
        if (threadIdx.x < st) {
            sS[threadIdx.x] += sS[threadIdx.x + st];
            sQ[threadIdx.x] += sQ[threadIdx.x + st];
        }
        __syncthreads();
    }
    if (threadIdx.x == 0) {
        part[2 * blockIdx.x]     = sS[0];
        part[2 * blockIdx.x + 1] = sQ[0];
    }
}

// ---------------------------------------------------------------------------
// K2: finalize BN1 -> (scale1, shift1). Deterministic fixed-order reduction.
// ---------------------------------------------------------------------------
__global__ __launch_bounds__(256) void k2_bn1fin(
    const float* __restrict__ part, const float* __restrict__ g1,
    const float* __restrict__ b1, float* __restrict__ prm)
{
    __shared__ float sS[256], sQ[256];
    float s = 0.f, q = 0.f;
    for (int i = threadIdx.x; i < 4096; i += 256) {
        s += part[2 * i];
        q += part[2 * i + 1];
    }
    sS[threadIdx.x] = s;
    sQ[threadIdx.x] = q;
    __syncthreads();
    for (int st = 128; st > 0; st >>= 1) {
        if (threadIdx.x < st) {
            sS[threadIdx.x] += sS[threadIdx.x + st];
            sQ[threadIdx.x] += sQ[threadIdx.x + st];
        }
        __syncthreads();
    }
    if (threadIdx.x == 0) {
        const float mean = sS[0] / (float)BVOX;
        const float var  = sQ[0] / (float)BVOX - mean * mean;
        const float sc   = g1[0] * rsqrtf(var + 1e-5f);
        prm[0] = sc;
        prm[1] = b1[0] - mean * sc;
    }
}

// ---------------------------------------------------------------------------
// conv2 as WMMA GEMM: D[32ch x 16vox] = W2[32x27] * Y[27x16], K padded to 28.
// V_WMMA_F32_16X16X4_F32, 7 K-steps x 2 channel tiles.
// A (16x4 f32): lanes 0-15 M=lane {K0,K1}; lanes 16-31 M=lane-16 {K2,K3}.
// B (4x16 f32) assumed mirror of C/D striping: VGPR r -> K=r (lanes 0-15) /
// K=r+2 (lanes 16-31), so A and B share per-lane k indices.
// B elements come from a zero-padded LDS tile of BN1-normalized values:
// inner loop is add + ds_load per element, no masks, no 64-bit addressing.
// ---------------------------------------------------------------------------
__device__ __forceinline__ constexpr int t_rel(int k) {
    const int kk = (k < 27) ? k : 26;    // k=27 pad aliases k=26 (A-weight 0)
    const int dz = kk / 9 - 1;
    const int dy = (kk / 3) % 3 - 1;
    const int dx = kk % 3 - 1;
    return dz * PS + dy * XS + dx;       // relative LDS offset (floats)
}

__device__ __forceinline__ void make_drel(bool hi, int drel[7][2])
{
    #pragma unroll
    for (int j = 0; j < 7; ++j) {
        #pragma unroll
        for (int e = 0; e < 2; ++e) {
            drel[j][e] = hi ? t_rel(4 * j + 2 + e) : t_rel(4 * j + e);
        }
    }
}

__device__ __forceinline__ void load_wfrag(const float* __restrict__ w2,
                                           int lane, v2f a0[7], v2f a1[7])
{
    const int m = lane & 15;
    const int khalf = (lane >= 16) ? 2 : 0;
    #pragma unroll
    for (int j = 0; j < 7; ++j) {
        const int k = 4 * j + khalf;
        a0[j].x = (k     < 27) ? w2[m * 27 + k]            : 0.f;
        a0[j].y = (k + 1 < 27) ? w2[m * 27 + k + 1]        : 0.f;
        a1[j].x = (k     < 27) ? w2[(16 + m) * 27 + k]     : 0.f;
        a1[j].y = (k + 1 < 27) ? w2[(16 + m) * 27 + k + 1] : 0.f;
    }
}

// Stage yn = scale1*fuzz + shift1 into LDS tile with zeroed halo.
// fzimg points PADF floats into ws: unmasked neighbor loads always in-bounds;
// invalid elements are zeroed by mask-multiply (loads stay unconditional).
__device__ __forceinline__ void stage_tile(float* __restrict__ tile,
    const float* __restrict__ fzimg, int d, int h0, float s1, float s0,
    int tid)
{
    const int pxb  = tid & 63;
    const int rowb = tid >> 6;           // 0..3
    #pragma unroll
    for (int pz = 0; pz < 3; ++pz) {
        const int dd = d + pz - 1;
        for (int py = rowb; py < 18; py += 4) {
            const int hh = h0 + py - 1;
            for (int px = pxb; px < 66; px += 64) {
                const int ww = px - 1;
                const bool v = ((unsigned)dd < 64u) & ((unsigned)hh < 64u) &
                               ((unsigned)ww < 64u);
                const float raw = fzimg[dd * HWSZ + hh * DIM + ww];
                const float mfl = v ? 1.f : 0.f;
                tile[pz * PS + py * XS + px] = fmaf(raw, s1, s0) * mfl;
            }
        }
    }
}

__device__ __forceinline__ void conv2_tile(const float* __restrict__ tile,
    int a0i, const int drel[7][2], const v2f a0[7], const v2f a1[7],
    v8f& c0, v8f& c1)
{
    float bx[7], by[7];
    #pragma unroll
    for (int j = 0; j < 7; ++j) {
        bx[j] = tile[a0i + drel[j][0]];
        by[j] = tile[a0i + drel[j][1]];
    }
    const v8f z = {0.f, 0.f, 0.f, 0.f, 0.f, 0.f, 0.f, 0.f};
    c0 = z;
    c1 = z;
    #pragma unroll
    for (int j = 0; j < 7; ++j) {
        v2f b;
        b.x = bx[j];
        b.y = by[j];
        c0 = __builtin_amdgcn_wmma_f32_16x16x4_f32(false, a0[j], false, b,
                                                   (short)0, c0, false, false);
        c1 = __builtin_amdgcn_wmma_f32_16x16x4_f32(false, a1[j], false, b,
                                                   (short)0, c1, false, false);
    }
}

// ---------------------------------------------------------------------------
// K3: conv2 via WMMA from LDS tile; per-channel (sum, sumsq) block partials.
// 1024 blocks; block = one (img, d, 16 h-rows) slab = 64 groups of 16 voxels.
// ---------------------------------------------------------------------------
__global__ __launch_bounds__(256) void k3_conv2_stats(
    const float* __restrict__ fuzz, const float* __restrict__ w2,
    const float* __restrict__ cb2, const float* __restrict__ prm1,
    float* __restrict__ part)
{
    __shared__ float tile[TILEF];
    __shared__ float sS[32], sQ[32];
    if (threadIdx.x < 32) { sS[threadIdx.x] = 0.f; sQ[threadIdx.x] = 0.f; }

    const float s1 = prm1[0], s0 = prm1[1];
    const int tid  = threadIdx.x;
    const int lane = tid & 31;
    const int wv   = tid >> 5;
    const bool hi  = lane >= 16;
    const int n    = lane & 15;

    const int g0   = blockIdx.x * 64;
    const int img  = g0 >> 14;
    const int rem0 = g0 & 16383;
    const int h0   = (rem0 >> 2) & 63;   // multiple of 16
    const int d    = rem0 >> 8;
    const float* fzimg = fuzz + (size_t)img * VOX;

    stage_tile(tile, fzimg, d, h0, s1, s0, tid);

    v2f a0[7], a1[7];
    load_wfrag(w2, lane, a0, a1);
    int drel[7][2];
    make_drel(hi, drel);

    const int mhi = hi ? 8 : 0;
    float bb0[8], bb1[8];
    #pragma unroll
    for (int r = 0; r < 8; ++r) {
        bb0[r] = cb2[r + mhi];
        bb1[r] = cb2[16 + r + mhi];
    }

    float lS[16], lQ[16];
    #pragma unroll
    for (int i = 0; i < 16; ++i) { lS[i] = 0.f; lQ[i] = 0.f; }

    __syncthreads();

    for (int i = 0; i < 8; ++i) {
        const int li   = wv * 8 + i;             // 0..63
        const int wg   = li & 3;
        const int hloc = li >> 2;
        const int a0i  = PS + (hloc + 1) * XS + (wg * 16 + n + 1);
        v8f c0, c1;
        conv2_tile(tile, a0i, drel, a0, a1, c0, c1);
        #pragma unroll
        for (int r = 0; r < 8; ++r) {
            const float v0 = c0[r] + bb0[r];
            const float v1 = c1[r] + bb1[r];
            lS[r]     += v0;  lQ[r]     += v0 * v0;
            lS[8 + r] += v1;  lQ[8 + r] += v1 * v1;
        }
    }
    #pragma unroll
    for (int r = 0; r < 8; ++r) {
        atomicAdd(&sS[r + mhi],      lS[r]);
        atomicAdd(&sQ[r + mhi],      lQ[r]);
        atomicAdd(&sS[16 + r + mhi], lS[8 + r]);
        atomicAdd(&sQ[16 + r + mhi], lQ[8 + r]);
    }
    __syncthreads();
    if (tid < 64) {
        part[blockIdx.x * 64 + tid] = (tid < 32) ? sS[tid] : sQ[tid - 32];
    }
}

// ---------------------------------------------------------------------------
// K3b: finalize BN2 -> scale2[32], shift2[32]
// ---------------------------------------------------------------------------
__global__ __launch_bounds__(64) void k3b_bn2fin(
    const float* __restrict__ part, const float* __restrict__ g2,
    const float* __restrict__ bb2, float* __restrict__ prm)
{
    __shared__ float red[64];
    const int tid = threadIdx.x;
    float s = 0.f;
    for (int b = 0; b < 1024; ++b) s += part[b * 64 + tid];
    red[tid] = s;
    __syncthreads();
    if (tid < 32) {
        const float S = red[tid], Q = red[32 + tid];
        const float mean = S / (float)BVOX;
        const float var  = Q / (float)BVOX - mean * mean;
        const float sc   = g2[tid] * rsqrtf(var + 1e-5f);
        prm[tid]      = sc;
        prm[32 + tid] = bb2[tid] - mean * sc;
    }
}

// ---------------------------------------------------------------------------
// K4: recompute conv2 via WMMA from LDS tile, apply BN2, stream output.
// Same block shape as K3.
// ---------------------------------------------------------------------------
__global__ __launch_bounds__(256) void k4_conv2_bn2_out(
    const float* __restrict__ fuzz, const float* __restrict__ w2,
    const float* __restrict__ cb2, const float* __restrict__ prm1,
    const float* __restrict__ prm2, float* __restrict__ out)
{
    __shared__ float tile[TILEF];

    const float s1 = prm1[0], s0 = prm1[1];
    const int tid  = threadIdx.x;
    const int lane = tid & 31;
    const int wv   = tid >> 5;
    const bool hi  = lane >= 16;
    const int n    = lane & 15;

    const int g0   = blockIdx.x * 64;
    const int img  = g0 >> 14;
    const int rem0 = g0 & 16383;
    const int h0   = (rem0 >> 2) & 63;
    const int d    = rem0 >> 8;
    const float* fzimg = fuzz + (size_t)img * VOX;

    stage_tile(tile, fzimg, d, h0, s1, s0, tid);

    v2f a0[7], a1[7];
    load_wfrag(w2, lane, a0, a1);
    int drel[7][2];
    make_drel(hi, drel);

    const int mhi = hi ? 8 : 0;
    float sc0[8], sh0[8], sc1[8], sh1[8], bb0[8], bb1[8];
    #pragma unroll
    for (int r = 0; r < 8; ++r) {
        sc0[r] = prm2[r + mhi];       sh0[r] = prm2[32 + r + mhi];
        sc1[r] = prm2[16 + r + mhi];  sh1[r] = prm2[48 + r + mhi];
        bb0[r] = cb2[r + mhi];        bb1[r] = cb2[16 + r + mhi];
    }

    __syncthreads();

    for (int i = 0; i < 8; ++i) {
        const int li   = wv * 8 + i;
        const int wg   = li & 3;
        const int hloc = li >> 2;
        const int w    = wg * 16 + n;
        const int a0i  = PS + (hloc + 1) * XS + (w + 1);

        v8f c0, c1;
        conv2_tile(tile, a0i, drel, a0, a1, c0, c1);

        const int vox = d * HWSZ + (h0 + hloc) * DIM + w;
        float* ob = out + (size_t)img * (CCH * VOX) + vox;
        #pragma unroll
        for (int r = 0; r < 8; ++r) {
            const float o0 = fmaf(c0[r] + bb0[r], sc0[r], sh0[r]);
            const float o1 = fmaf(c1[r] + bb1[r], sc1[r], sh1[r]);
            __builtin_nontemporal_store(o0, ob + (size_t)(r + mhi) * VOX);
            __builtin_nontemporal_store(o1, ob + (size_t)(16 + r + mhi) * VOX);
        }
    }
}

// ---------------------------------------------------------------------------
extern "C" void kernel_launch(void* const* d_in, const int* in_sizes, int n_in,
                              void* d_out, int out_size, void* d_ws, size_t ws_size,
                              hipStream_t stream)
{
    (void)in_sizes; (void)n_in; (void)out_size; (void)ws_size;
    const float* x   = (const float*)d_in[0];
    const float* w1  = (const float*)d_in[1];
    const float* cb1 = (const float*)d_in[2];
    const float* w2  = (const float*)d_in[3];
    const float* cb2 = (const float*)d_in[4];
    const float* mu  = (const float*)d_in[5];
    const float* sg  = (const float*)d_in[6];
    const float* g1  = (const float*)d_in[7];
    const float* b1  = (const float*)d_in[8];
    const float* g2  = (const float*)d_in[9];
    const float* b2  = (const float*)d_in[10];
    float* out = (float*)d_out;
    float* ws  = (float*)d_ws;

    float* fuzz = ws + WS_FUZZ + PADF;   // padded on both sides
    float* p1   = ws + WS_P1;
    float* prm1 = ws + WS_BN1;
    float* p2   = ws + WS_P2;
    float* prm2 = ws + WS_BN2;

    k1_conv1_fuzz   <<<4096, 256, 0, stream>>>(x, w1, cb1, mu, sg, fuzz, p1);
    k2_bn1fin       <<<1,    256, 0, stream>>>(p1, g1, b1, prm1);
    k3_conv2_stats  <<<1024, 256, 0, stream>>>(fuzz, w2, cb2, prm1, p2);
    k3b_bn2fin      <<<1,     64, 0, stream>>>(p2, g2, b2, prm2);
    k4_conv2_bn2_out<<<1024, 256, 0, stream>>>(fuzz, w2, cb2, prm1, prm2, out);
}